// EmbGCN2_3470333575180
// MI455X (gfx1250) — compile-verified
//
#include <hip/hip_runtime.h>
#include <hip/hip_bf16.h>
#include <math.h>

// Problem constants (match reference)
#define NN 5000
#define BB 64
#define CI 32
#define CO 32
#define DD 10

typedef __attribute__((ext_vector_type(2))) float v2f;
typedef __attribute__((ext_vector_type(8))) float v8f;

// D = A(16x4 f32) * B(4x16 f32) + C(16x16 f32)   -- CDNA5 fp32 WMMA
__device__ __forceinline__ v8f wmma_f32x4(v2f a, v2f b, v8f c) {
  return __builtin_amdgcn_wmma_f32_16x16x4_f32(
      /*neg_a=*/false, a, /*neg_b=*/false, b,
      /*c_mod=*/(short)0, c, /*reuse_a=*/false, /*reuse_b=*/false);
}

__device__ __forceinline__ v8f v8f_zero() {
  v8f z = {0.f, 0.f, 0.f, 0.f, 0.f, 0.f, 0.f, 0.f};
  return z;
}

// ---------------- block reductions (256 threads) ----------------
__device__ __forceinline__ float blockReduceMax256(float v, float* red) {
  int t = threadIdx.x;
  red[t] = v;
  __syncthreads();
  for (int s = 128; s > 0; s >>= 1) {
    if (t < s) red[t] = fmaxf(red[t], red[t + s]);
    __syncthreads();
  }
  float r = red[0];
  __syncthreads();
  return r;
}

__device__ __forceinline__ float blockReduceSum256(float v, float* red) {
  int t = threadIdx.x;
  red[t] = v;
  __syncthreads();
  for (int s = 128; s > 0; s >>= 1) {
    if (t < s) red[t] = red[t] + red[t + s];
    __syncthreads();
  }
  float r = red[0];
  __syncthreads();
  return r;
}

// ---------------- K0: zero init ----------------
__global__ void zero_kernel(float* __restrict__ p, int n) {
  int i = blockIdx.x * 256 + threadIdx.x;
  if (i < n) p[i] = 0.f;
}

// ---------------- K0b: pack x into k-pair-interleaved layout ----------------
// xp[((b*2500 + m/2)*32 + c)*2 + (m&1)] = x[b,m,c]
// Makes each WMMA B-fragment (two consecutive k rows at one column) a single
// aligned 8-byte load.
__global__ __launch_bounds__(256) void pack_x_kernel(
    const float* __restrict__ x, float* __restrict__ xp) {
  size_t i = (size_t)blockIdx.x * 256 + threadIdx.x;  // over B*N*CI
  if (i >= (size_t)BB * NN * CI) return;
  int r = (int)(i & 1);
  size_t q = i >> 1;
  int c = (int)(q & 31);
  q >>= 5;
  int mp = (int)(q % (NN / 2));
  int b = (int)(q / (NN / 2));
  int m = mp * 2 + r;
  xp[i] = x[((size_t)b * NN + m) * CI + c];
}

// ---------------- K1: supports = softmax(relu(E E^T), rows) + I ----------------
__global__ __launch_bounds__(256) void supports_kernel(
    const float* __restrict__ E, float* __restrict__ S) {
  __shared__ float EnS[16];
  __shared__ float red[256];
  int n = blockIdx.x;
  int t = threadIdx.x;
  if (t < DD) EnS[t] = E[(size_t)n * DD + t];
  __syncthreads();
  float en[DD];
#pragma unroll
  for (int d = 0; d < DD; ++d) en[d] = EnS[d];

  float lmax = 0.f;  // relu >= 0
  for (int m = t; m < NN; m += 256) {
    const float* Em = E + (size_t)m * DD;
    float v = 0.f;
#pragma unroll
    for (int d = 0; d < DD; ++d) v += en[d] * Em[d];
    v = fmaxf(v, 0.f);
    lmax = fmaxf(lmax, v);
  }
  float rowmax = blockReduceMax256(lmax, red);

  float lsum = 0.f;
  for (int m = t; m < NN; m += 256) {
    const float* Em = E + (size_t)m * DD;
    float v = 0.f;
#pragma unroll
    for (int d = 0; d < DD; ++d) v += en[d] * Em[d];
    v = fmaxf(v, 0.f);
    lsum += __expf(v - rowmax);
  }
  float inv = 1.f / blockReduceSum256(lsum, red);

  for (int m = t; m < NN; m += 256) {
    const float* Em = E + (size_t)m * DD;
    float v = 0.f;
#pragma unroll
    for (int d = 0; d < DD; ++d) v += en[d] * Em[d];
    v = fmaxf(v, 0.f);
    float e = __expf(v - rowmax) * inv;
    if (m == n) e += 1.f;
    S[(size_t)n * NN + m] = e;
  }
}

// ---------------- K2: col[m] = sum_n softmax(sym_adj, rows)[n, m] ----------------
#define ROWS2 25
__global__ __launch_bounds__(256) void col_kernel(
    const float* __restrict__ A, float* __restrict__ colg) {
  __shared__ float cp[NN];    // per-block column partials (thread-owned stripes)
  __shared__ float red[256];
  int t = threadIdx.x;
  for (int m = t; m < NN; m += 256) cp[m] = 0.f;
  __syncthreads();

  int r0 = blockIdx.x * ROWS2;  // 200 blocks * 25 rows = 5000
  for (int r = 0; r < ROWS2; ++r) {
    int row = r0 + r;
    const float* Ar = A + (size_t)row * NN;
    float lmax = -3.4e38f;
    for (int m = t; m < NN; m += 256) lmax = fmaxf(lmax, Ar[m]);
    float rowmax = blockReduceMax256(lmax, red);
    float lsum = 0.f;
    for (int m = t; m < NN; m += 256) lsum += __expf(Ar[m] - rowmax);
    float inv = 1.f / blockReduceSum256(lsum, red);
    for (int m = t; m < NN; m += 256) cp[m] += __expf(Ar[m] - rowmax) * inv;
  }
  __syncthreads();
  for (int m = t; m < NN; m += 256) atomicAdd(&colg[m], cp[m]);
}

// ---------------- K3: per-node weights/bias from pools ----------------
__global__ __launch_bounds__(256) void genw_kernel(
    const float* __restrict__ E, const float* __restrict__ Wp,
    const float* __restrict__ bp, float* __restrict__ Wn,
    float* __restrict__ bn) {
  __shared__ float EnS[16];
  int n = blockIdx.x;
  int t = threadIdx.x;
  if (t < DD) EnS[t] = E[(size_t)n * DD + t];
  __syncthreads();
  float en[DD];
#pragma unroll
  for (int d = 0; d < DD; ++d) en[d] = EnS[d];
#pragma unroll
  for (int j = 0; j < 4; ++j) {
    int idx = t + j * 256;  // 0..1023 over [CI][CO]
    float acc = 0.f;
#pragma unroll
    for (int d = 0; d < DD; ++d) acc += en[d] * Wp[d * (CI * CO) + idx];
    Wn[(size_t)n * (CI * CO) + idx] = acc;
  }
  if (t < CO) {
    float acc = 0.f;
#pragma unroll
    for (int d = 0; d < DD; ++d) acc += en[d] * bp[d * CO + t];
    bn[(size_t)n * CO + t] = acc;
  }
}

// ---------------- K4: x_g[b,n,c] = sum_m S[n,m] x[b,m,c]  (fp32 WMMA GEMM) ----
// Block: 64-row S panel (4 n-tiles) staged in LDS. 8 waves; wave slot
// (blockIdx.y*8 + wave) covers (ctile in {0,1}) x (batch pair in 0..31).
// Each wave holds 4 ntile x 2 batch = 8 accumulators, so each B fragment
// (one global_load_b64 thanks to packed xp) feeds 4 WMMAs and each A fragment
// (ds_load_b64) feeds 2.
#define TK 64
__global__ __launch_bounds__(256) void gemm_supports_kernel(
    const float* __restrict__ S, const float* __restrict__ xp,
    float* __restrict__ xg) {
  __shared__ float s_lds[64][TK + 4];  // row stride 68 floats: bank stride 4

  int n0 = blockIdx.x * 64;
  int wave = threadIdx.x >> 5;
  int lane = threadIdx.x & 31;
  int slot = blockIdx.y * 8 + wave;  // 0..63
  int ctile = slot & 1;
  int b0 = (slot >> 1) * 2;          // 2 batches per wave
  int c0 = ctile * 16;

  int mrow = lane & 15;              // A row within tile / B-D column in tile
  int khalf = (lane >> 4) * 2;       // k sub-position per ISA 32-bit layout
  int cc = c0 + mrow;

  v8f acc[4][2];
#pragma unroll
  for (int j = 0; j < 4; ++j)
#pragma unroll
    for (int i = 0; i < 2; ++i) acc[j][i] = v8f_zero();

  for (int k0 = 0; k0 < NN; k0 += TK) {
    int klen = NN - k0;
    if (klen > TK) klen = TK;
    // cooperative coalesced load of S[n0..n0+63, k0..k0+klen) as float4
#pragma unroll
    for (int i = 0; i < 4; ++i) {
      int f = threadIdx.x + i * 256;   // 0..1023 float4 slots
      int row = f >> 4;                // 0..63
      int c4 = (f & 15) * 4;           // 0..60
      float4 v = {0.f, 0.f, 0.f, 0.f};
      int gr = n0 + row;
      if (gr < NN && c4 < klen)
        v = *(const float4*)(S + (size_t)gr * NN + k0 + c4);
      *(float4*)&s_lds[row][c4] = v;
    }
    __syncthreads();

    int ksteps = klen >> 2;
    for (int kk = 0; kk < ksteps; ++kk) {
      int kb = kk * 4 + khalf;
      int mg = k0 + kb;  // even
      v2f a[4];
#pragma unroll
      for (int j = 0; j < 4; ++j)
        a[j] = *(const v2f*)&s_lds[j * 16 + mrow][kb];
      v2f bv[2];
#pragma unroll
      for (int i = 0; i < 2; ++i)
        bv[i] = *(const v2f*)(xp +
            ((((size_t)(b0 + i) * (NN / 2) + (mg >> 1)) * CI + cc) << 1));
#pragma unroll
      for (int j = 0; j < 4; ++j)
#pragma unroll
        for (int i = 0; i < 2; ++i)
          acc[j][i] = wmma_f32x4(a[j], bv[i], acc[j][i]);
    }
    __syncthreads();
  }

  // store D: element (r, lane) -> row n0 + j*16 + r + 8*(lane>>4), col cc
  int half = 8 * (lane >> 4);
#pragma unroll
  for (int j = 0; j < 4; ++j) {
#pragma unroll
    for (int i = 0; i < 2; ++i) {
#pragma unroll
      for (int r = 0; r < 8; ++r) {
        int nn = n0 + j * 16 + r + half;
        if (nn < NN)
          xg[((size_t)(b0 + i) * NN + nn) * CI + cc] = acc[j][i][r];
      }
    }
  }
}

// ---------------- K5: per-node dual GEMM [64x32]@[32x32] + epilogue ----------
// out[b,n,o] = (x_g[b,n,:] @ W_n)[o] + bias_n[o] + sigmoid(xs)*xs,
//   xs = (col[n]*x[b,n,:] @ lin_w^T)[o] + lin_b[o]
__global__ __launch_bounds__(256) void final_kernel(
    const float* __restrict__ xg, const float* __restrict__ x,
    const float* __restrict__ Wn, const float* __restrict__ bn,
    const float* __restrict__ colv, const float* __restrict__ lin_w,
    const float* __restrict__ lin_b, float* __restrict__ out) {
  __shared__ float ag[BB][CI + 1];    // x_g[:, n, :]
  __shared__ float as[BB][CI + 1];    // col[n] * x[:, n, :]
  __shared__ float wg[CI][CO + 1];    // W_n
  __shared__ float wst[CI][CO + 1];   // lin_w transposed: [i][o]
  __shared__ float bn_l[CO];
  __shared__ float lb_l[CO];

  int n = blockIdx.x;
  int t = threadIdx.x;
  float cv = colv[n];

#pragma unroll
  for (int i = 0; i < 8; ++i) {
    int idx = t + i * 256;          // 0..2047 over [b][c]
    int b = idx >> 5, c = idx & 31;
    size_t g = ((size_t)b * NN + n) * CI + c;
    ag[b][c] = xg[g];
    as[b][c] = cv * x[g];
  }
#pragma unroll
  for (int i = 0; i < 4; ++i) {
    int idx = t + i * 256;          // 0..1023 over [i][o]
    int ii = idx >> 5, o = idx & 31;
    wg[ii][o] = Wn[(size_t)n * (CI * CO) + idx];
    wst[ii][o] = lin_w[o * CI + ii];
  }
  if (t < CO) {
    bn_l[t] = bn[(size_t)n * CO + t];
    lb_l[t] = lin_b[t];
  }
  __syncthreads();

  int wave = t >> 5, lane = t & 31;
  int mtile = wave >> 1, otile = wave & 1;  // 4 b-tiles x 2 o-tiles
  int mr = lane & 15;
  int kh = (lane >> 4) * 2;
  int brow = mtile * 16 + mr;
  int ocol = otile * 16 + mr;

  v8f accg = v8f_zero(), accs = v8f_zero();
#pragma unroll
  for (int kk = 0; kk < 8; ++kk) {
    int kb = kk * 4 + kh;
    v2f a1, b1, a2, b2;
    a1.x = ag[brow][kb];  a1.y = ag[brow][kb + 1];
    b1.x = wg[kb][ocol];  b1.y = wg[kb + 1][ocol];
    accg = wmma_f32x4(a1, b1, accg);
    a2.x = as[brow][kb];  a2.y = as[brow][kb + 1];
    b2.x = wst[kb][ocol]; b2.y = wst[kb + 1][ocol];
    accs = wmma_f32x4(a2, b2, accs);
  }

  int half = 8 * (lane >> 4);
  int o = otile * 16 + (lane & 15);
#pragma unroll
  for (int r = 0; r < 8; ++r) {
    int b = mtile * 16 + r + half;
    float g = accg[r] + bn_l[o];
    float xs = accs[r] + lb_l[o];
    float sg = 1.f / (1.f + __expf(-xs));
    out[((size_t)b * NN + n) * CO + o] = g + sg * xs;
  }
}

// ---------------- host launch ----------------
extern "C" void kernel_launch(void* const* d_in, const int* in_sizes, int n_in,
                              void* d_out, int out_size, void* d_ws,
                              size_t ws_size, hipStream_t stream) {
  const float* x     = (const float*)d_in[0];  // [B,N,CI]
  const float* E     = (const float*)d_in[1];  // [N,D]
  const float* A     = (const float*)d_in[2];  // [N,N]
  const float* lin_w = (const float*)d_in[3];  // [CO,CI]
  const float* lin_b = (const float*)d_in[4];  // [CO]
  const float* Wp    = (const float*)d_in[5];  // [D,CI,CO]
  const float* bp    = (const float*)d_in[6];  // [D,CO]
  float* out = (float*)d_out;

  // workspace layout (floats)
  float* S    = (float*)d_ws;                       // N*N
  float* xg   = S    + (size_t)NN * NN;             // B*N*CI
  float* xpk  = xg   + (size_t)BB * NN * CI;        // B*N*CI (packed x)
  float* Wn   = xpk  + (size_t)BB * NN * CI;        // N*CI*CO
  float* bn   = Wn   + (size_t)NN * CI * CO;        // N*CO
  float* colv = bn   + (size_t)NN * CO;             // N

  zero_kernel<<<(NN + 255) / 256, 256, 0, stream>>>(colv, NN);
  pack_x_kernel<<<(int)(((size_t)BB * NN * CI + 255) / 256), 256, 0, stream>>>(
      x, xpk);
  supports_kernel<<<NN, 256, 0, stream>>>(E, S);
  col_kernel<<<NN / ROWS2, 256, 0, stream>>>(A, colv);
  genw_kernel<<<NN, 256, 0, stream>>>(E, Wp, bp, Wn, bn);

  dim3 g4((NN + 63) / 64, 8);
  gemm_supports_kernel<<<g4, 256, 0, stream>>>(S, xpk, xg);
  final_kernel<<<NN, 256, 0, stream>>>(xg, x, Wn, bn, colv, lin_w, lin_b, out);
}